// Transformation_71270687309925
// MI455X (gfx1250) — compile-verified
//
#include <hip/hip_runtime.h>
#include <cstdint>

// ---------------------------------------------------------------------------
// Monotone cubic B-spline eval + log-derivative, CDNA5 / gfx1250.
//  N=500000 samples x V=8 vars, degree=40 => M=42 coefs, K=46 uniform knots.
//  d_out = [ y (N*V) | log_dy (N*V) ]  (float32)
//
// Memory roofline: 16 MB in + 32 MB out = 48 MB  =>  ~2 us at 23.3 TB/s.
// Hot kernel: 4 elements/thread (B128 in, 2x B128 out), de Boor reduced to
// 9 FMAs/elem in knot-space (zero divides), v_log_f32 for log(dy).
// Shared spline tables (2.7 KB) staged per-block into LDS via the CDNA5
// async path: global_load_async_to_lds_b128 + s_wait_asynccnt.
// ---------------------------------------------------------------------------

#define NVARS   8
#define MCOEF   42      // number of spline coefficients
#define NKNOT   46      // number of knots
// workspace / LDS layout (floats)
#define OFF_C   0                         // c  : 42*8 = 336
#define OFF_Q   (OFF_C + MCOEF*NVARS)     // q  : 41*8 = 328  -> 336..663
#define OFF_T0  (OFF_Q + (MCOEF-1)*NVARS) // t0 : 8           -> 664..671 (16B-aligned)
#define OFF_ID  (OFF_T0 + NVARS)          // 1/d: 8           -> 672..679 (16B-aligned)
#define TBL_TOTAL (OFF_ID + NVARS)        // 680 floats = 2720 B (170 x b128)

// ---------------------------------------------------------------------------
// Kernel 1: tiny precompute (one 64-thread block).  Thread v (<8) owns variable v:
//   incr = [raw0, softplus(raw1..)] ; c = cumsum(incr)
//   q[m] = 3*(c[m+1]-c[m]) / (t[m+4]-t[m+1]) ; plus t0, 1/dknot (uniform knots)
// ---------------------------------------------------------------------------
__global__ void spline_precompute(const float* __restrict__ raw,
                                  const float* __restrict__ knots,
                                  float* __restrict__ ws) {
  int v = threadIdx.x;
  if (v >= NVARS) return;

  float t0 = knots[v];
  float dk = knots[NVARS + v] - t0;
  ws[OFF_T0 + v] = t0;
  ws[OFF_ID + v] = 1.0f / dk;

  float acc = raw[v];
  ws[OFF_C + v] = acc;
  for (int m = 1; m < MCOEF; ++m) {
    float r  = raw[m * NVARS + v];
    float sp = fmaxf(r, 0.0f) + log1pf(expf(-fabsf(r)));   // stable softplus
    acc += sp;
    ws[OFF_C + m * NVARS + v] = acc;
  }
  for (int m = 0; m < MCOEF - 1; ++m) {
    float cm  = ws[OFF_C + m       * NVARS + v];
    float cm1 = ws[OFF_C + (m + 1) * NVARS + v];
    float den = knots[(m + 4) * NVARS + v] - knots[(m + 1) * NVARS + v];
    ws[OFF_Q + m * NVARS + v] = 3.0f * (cm1 - cm) / den;
  }
}

// ---------------------------------------------------------------------------
// Per-element evaluation: u = (x-t0)/d ; k = clip(floor(u),3,41);
// all de Boor denominators collapse to {d,2d,3d} for uniform knots.
// ---------------------------------------------------------------------------
__device__ __forceinline__ void eval_elem(const float* __restrict__ tbl,
                                          float xx, float t0, float invd, int v,
                                          float& yout, float& ldout) {
  float u = (xx - t0) * invd;
  int k = (int)floorf(u);
  k = max(3, min(NKNOT - 5, k));          // [3, 41]
  float uk = u - (float)k;

  const float* cb = &tbl[OFF_C + (k - 3) * NVARS + v];
  float d0 = cb[0], d1 = cb[8], d2 = cb[16], d3 = cb[24];
  const float* qb = &tbl[OFF_Q + (k - 3) * NVARS + v];
  float e0 = qb[0], e1 = qb[8], e2 = qb[16];

  const float THIRD = 0.3333333333333333f;
  float a;
  // cubic de Boor (p=3): alpha = (u-k+s)/(4-r)
  a = uk * THIRD;           d3 = fmaf(a, d3 - d2, d2);   // r=1, jj=3
  a = (uk + 1.f) * THIRD;   d2 = fmaf(a, d2 - d1, d1);   // r=1, jj=2
  a = (uk + 2.f) * THIRD;   d1 = fmaf(a, d1 - d0, d0);   // r=1, jj=1
  a = uk * 0.5f;            d3 = fmaf(a, d3 - d2, d2);   // r=2, jj=3
  a = (uk + 1.f) * 0.5f;    d2 = fmaf(a, d2 - d1, d1);   // r=2, jj=2
  a = uk;                   d3 = fmaf(a, d3 - d2, d2);   // r=3, jj=3
  // derivative spline (p=2, interval k-1): alpha = (u-k+s)/(3-r)
  a = uk * 0.5f;            e2 = fmaf(a, e2 - e1, e1);   // r=1, jj=2
  a = (uk + 1.f) * 0.5f;    e1 = fmaf(a, e1 - e0, e0);   // r=1, jj=1
  a = uk;                   e2 = fmaf(a, e2 - e1, e1);   // r=2, jj=2

  yout = d3;
  ldout = __logf(e2);                     // v_log_f32 path
}

// ---------------------------------------------------------------------------
// Kernel 2: evaluation, 4 elements per thread.
// base%4==0 => the 4 variables are vbase..vbase+3 (vbase in {0,4}), so t0/invd
// come from two contiguous ds_load_b128s; x/y/log_dy are B128 per lane.
// ---------------------------------------------------------------------------
__global__ __launch_bounds__(256) void spline_eval(
    const float* __restrict__ x, const float* __restrict__ ws,
    float* __restrict__ y, float* __restrict__ logd, int total) {

  __shared__ __align__(16) float tbl[TBL_TOTAL];

  // --- CDNA5 async stage of the 2720-B table into LDS (170 b128 xfers) ----
  int tid = threadIdx.x;
  if (tid < TBL_TOTAL / 4) {
    unsigned lds_addr = (unsigned)(uintptr_t)(&tbl[tid * 4]);  // low 32b of flat ptr = LDS offset
    unsigned long long gaddr = (unsigned long long)(uintptr_t)(ws + tid * 4);
    asm volatile("global_load_async_to_lds_b128 %0, %1, off"
                 :: "v"(lds_addr), "v"(gaddr) : "memory");
  }
#if __has_builtin(__builtin_amdgcn_s_wait_asynccnt)
  __builtin_amdgcn_s_wait_asynccnt(0);
#else
  asm volatile("s_wait_asynccnt 0" ::: "memory");
#endif
  __syncthreads();

  int base = (blockIdx.x * 256 + tid) * 4;
  if (base >= total) return;
  int vbase = base & (NVARS - 1);         // 0 or 4

  float4 t04  = *(const float4*)&tbl[OFF_T0 + vbase];
  float4 id4  = *(const float4*)&tbl[OFF_ID + vbase];

  if (base + 3 < total) {
    float4 xv = *(const float4*)(x + base);
    float4 yv, lv;
    eval_elem(tbl, xv.x, t04.x, id4.x, vbase + 0, yv.x, lv.x);
    eval_elem(tbl, xv.y, t04.y, id4.y, vbase + 1, yv.y, lv.y);
    eval_elem(tbl, xv.z, t04.z, id4.z, vbase + 2, yv.z, lv.z);
    eval_elem(tbl, xv.w, t04.w, id4.w, vbase + 3, yv.w, lv.w);
    *(float4*)(y + base)    = yv;
    *(float4*)(logd + base) = lv;
  } else {
    const float t0s[4] = {t04.x, t04.y, t04.z, t04.w};
    const float ids[4] = {id4.x, id4.y, id4.z, id4.w};
    for (int j = 0; j < 4; ++j) {
      int g = base + j;
      if (g >= total) break;
      float yo, lo;
      eval_elem(tbl, x[g], t0s[j], ids[j], vbase + j, yo, lo);
      y[g] = yo;
      logd[g] = lo;
    }
  }
}

// ---------------------------------------------------------------------------
extern "C" void kernel_launch(void* const* d_in, const int* in_sizes, int n_in,
                              void* d_out, int out_size, void* d_ws, size_t ws_size,
                              hipStream_t stream) {
  const float* x     = (const float*)d_in[0];   // (N, 8)
  const float* raw   = (const float*)d_in[1];   // (42, 8)
  const float* knots = (const float*)d_in[2];   // (46, 8)
  float* ws   = (float*)d_ws;
  float* y    = (float*)d_out;
  int total   = in_sizes[0];                    // N*V = 4,000,000
  float* logd = y + total;

  spline_precompute<<<1, 64, 0, stream>>>(raw, knots, ws);
  int threads_needed = (total + 3) / 4;
  int blocks = (threads_needed + 255) / 256;
  spline_eval<<<blocks, 256, 0, stream>>>(x, ws, y, logd, total);
}